// model_memory_IRM_45423574123201
// MI455X (gfx1250) — compile-verified
//
#include <hip/hip_runtime.h>
#include <math.h>

#define B_ 512
#define M_ 32768
#define PAST_ 8
#define FUT_ 12
#define NPRED_ 20
#define D_ 48
#define NROW (B_*NPRED_)     // 10240
#define NCHUNK 64
#define CHUNK (M_/NCHUNK)    // 512
#define TILES (CHUNK/16)     // 32
#define LOG2E_ 1.4426950408889634f

typedef __attribute__((ext_vector_type(2))) float v2f;
typedef __attribute__((ext_vector_type(8))) float v8f;

__device__ __forceinline__ float sigmoidf_(float x){ return 1.0f/(1.0f + __expf(-x)); }
// raw v_exp_f32: flushes denormal results, which are numerically irrelevant for softmax weights
__device__ __forceinline__ float fast_exp2_(float x){ return __builtin_amdgcn_exp2f(x); }

// ---------------- encoder: Conv1d(2->48,k3,pad1)+ReLU then GRU(48) over T=8 ----------------
__global__ __launch_bounds__(64)
void enc_kernel(const float* __restrict__ past, const float* __restrict__ conv_w,
                const float* __restrict__ conv_b,
                const float* __restrict__ w_ih, const float* __restrict__ w_hh,
                const float* __restrict__ b_ih, const float* __restrict__ b_hh,
                float* __restrict__ state_past)
{
    int b = blockIdx.x, tid = threadIdx.x;
    __shared__ float xl[PAST_][2];
    __shared__ float emb[PAST_][D_];
    __shared__ float h[D_];
    if (tid < PAST_*2) xl[tid>>1][tid&1] = past[b*PAST_*2 + tid];
    if (tid < D_) h[tid] = 0.0f;
    __syncthreads();
    if (tid < D_) {
        float w[6];
        #pragma unroll
        for (int i=0;i<6;i++) w[i] = conv_w[tid*6 + i];
        float bc = conv_b[tid];
        for (int t=0;t<PAST_;t++){
            float s = bc;
            #pragma unroll
            for (int k=0;k<3;k++){
                int tt = t-1+k;
                if (tt>=0 && tt<PAST_)
                    s += xl[tt][0]*w[0*3+k] + xl[tt][1]*w[1*3+k];
            }
            emb[t][tid] = fmaxf(s, 0.0f);
        }
    }
    __syncthreads();
    for (int t=0;t<PAST_;t++){
        float hn = 0.0f;
        if (tid < D_){
            float gi[3], gh[3];
            #pragma unroll
            for (int g=0; g<3; g++){
                const float* wi = w_ih + (g*D_+tid)*D_;
                const float* wh = w_hh + (g*D_+tid)*D_;
                float si = b_ih[g*D_+tid], sh = b_hh[g*D_+tid];
                for (int k=0;k<D_;k++){ si += emb[t][k]*wi[k]; sh += h[k]*wh[k]; }
                gi[g]=si; gh[g]=sh;
            }
            float r = sigmoidf_(gi[0]+gh[0]);
            float z = sigmoidf_(gi[1]+gh[1]);
            float nn = tanhf(gi[2] + r*gh[2]);
            hn = (1.0f-z)*nn + z*h[tid];
        }
        __syncthreads();
        if (tid < D_) h[tid] = hn;
        __syncthreads();
    }
    if (tid < D_) state_past[b*D_+tid] = h[tid];
}

// -------- K/V projection, written PRE-SWIZZLED into per-lane WMMA A-operand order ----------
// K: scores A-operand. lane = (m%16) + 16*h, d = 4s+2h+j            -> slot o = 2s+j
// V: P@V    A-operand. lane = (d%16) + 16*h, m%16 = 4s+2h+j, t=d/16 -> slot o = (s*3+t)*2+j
// Each (tile,lane) owns 24 contiguous floats (6 x b128); every WMMA operand pair is a .xy/.zw
// half of a loaded float4 -> no register shuffles in the hot loop.
__global__ void kv_kernel(const float* __restrict__ mp, const float* __restrict__ mf,
                          const float* __restrict__ Wk, const float* __restrict__ bk,
                          const float* __restrict__ Wv, const float* __restrict__ bv,
                          float* __restrict__ Kswz, float* __restrict__ Vswz){
    int idx = blockIdx.x*blockDim.x + threadIdx.x;
    if (idx >= M_*D_) return;
    int m = idx / D_, d = idx % D_;
    const float* pr = mp + (size_t)m*D_; const float* fr = mf + (size_t)m*D_;
    const float* wk = Wk + d*D_;         const float* wv = Wv + d*D_;
    float sk = bk[d], sv = bv[d];
    for (int k=0;k<D_;k++){ sk += pr[k]*wk[k]; sv += fr[k]*wv[k]; }
    int tile = m >> 4;
    // K swizzle
    {
        int col = m & 15;
        int s = d >> 2, h = (d >> 1) & 1, j = d & 1;
        int lane = col + 16*h;
        Kswz[(size_t)tile*768 + lane*24 + 2*s + j] = sk;
    }
    // V swizzle
    {
        int mloc = m & 15;
        int s = mloc >> 2, h = (mloc >> 1) & 1, j = mloc & 1;
        int t = d >> 4, col = d & 15;
        int lane = col + 16*h;
        Vswz[(size_t)tile*768 + lane*24 + (s*3 + t)*2 + j] = sv;
    }
}

// ---------------- q = c@Wq^T + bq ----------------
__global__ void q_kernel(const float* __restrict__ c, const float* __restrict__ Wq,
                         const float* __restrict__ bq, float* __restrict__ q){
    int idx = blockIdx.x*blockDim.x + threadIdx.x;
    if (idx >= B_*D_) return;
    int b = idx / D_, d = idx % D_;
    const float* cr = c + b*D_; const float* w = Wq + d*D_;
    float s = bq[d];
    for (int k=0;k<D_;k++) s += cr[k]*w[k];
    q[idx] = s;
}

// ---- helpers for the flash-attention hot loop (double-buffered software pipeline) ----
__device__ __forceinline__ void attn_tile_load(const float* kb, const float* vb,
                                               float (&ka)[24], float (&va)[24])
{
    const float4* kp = (const float4*)kb;
    const float4* vp = (const float4*)vb;
    #pragma unroll
    for (int i=0;i<6;i++){
        float4 f = kp[i]; ka[4*i]=f.x; ka[4*i+1]=f.y; ka[4*i+2]=f.z; ka[4*i+3]=f.w;
        float4 g = vp[i]; va[4*i]=g.x; va[4*i+1]=g.y; va[4*i+2]=g.z; va[4*i+3]=g.w;
    }
}

__device__ __forceinline__ void attn_tile_compute(const float (&ka)[24], const float (&va)[24],
                                                  const v2f (&qb)[12],
                                                  v8f &acc0, v8f &acc1, v8f &acc2,
                                                  float &mx, float &mxL, float &sm, int half)
{
    // ---- S^T tile: two independent WMMA accumulation chains (depth 6) ----
    v8f Sa = {0,0,0,0,0,0,0,0}, Sb = Sa;
    #pragma unroll
    for (int s=0;s<12;s+=2){
        v2f a0 = (v2f){ka[2*s],   ka[2*s+1]};
        v2f a1 = (v2f){ka[2*s+2], ka[2*s+3]};
        Sa = __builtin_amdgcn_wmma_f32_16x16x4_f32(false, a0, false, qb[s],   (short)0, Sa, false, false);
        Sb = __builtin_amdgcn_wmma_f32_16x16x4_f32(false, a1, false, qb[s+1], (short)0, Sb, false, false);
    }
    v8f Sc;
    #pragma unroll
    for (int j=0;j<8;j++) Sc[j] = Sa[j] + Sb[j];

    // ---- convert raw scores into P^T B-operand layout (overlaps max reduction) ----
    float sw[8];
    #pragma unroll
    for (int j=0;j<8;j++) sw[j] = __shfl_xor(Sc[j], 16, 32);
    float r[8];
    r[0] = half ? sw[2] : Sc[0];  r[1] = half ? sw[3] : Sc[1];   // s=0
    r[2] = half ? sw[6] : Sc[4];  r[3] = half ? sw[7] : Sc[5];   // s=1
    r[4] = half ? Sc[2] : sw[0];  r[5] = half ? Sc[3] : sw[1];   // s=2
    r[6] = half ? Sc[6] : sw[4];  r[7] = half ? Sc[7] : sw[5];   // s=3

    // ---- online softmax (per batch column; halves hold complementary m-subsets) ----
    float t01 = fmaxf(r[0],r[1]), t23 = fmaxf(r[2],r[3]);
    float t45 = fmaxf(r[4],r[5]), t67 = fmaxf(r[6],r[7]);
    float tmax = fmaxf(fmaxf(t01,t23), fmaxf(t45,t67));
    tmax = fmaxf(tmax, __shfl_xor(tmax, 16, 32));
    if (__any(tmax > mx)) {                   // wave-uniform branch: EXEC stays all-ones
        float nm  = fmaxf(mx, tmax);
        float nmL = nm * LOG2E_;
        float alpha = fast_exp2_(mxL - nmL);  // lane-local rescale (acc columns are b)
        sm *= alpha;
        #pragma unroll
        for (int j=0;j<8;j++){ acc0[j]*=alpha; acc1[j]*=alpha; acc2[j]*=alpha; }
        mx = nm; mxL = nmL;
    }
    float p[8];
    #pragma unroll
    for (int i=0;i<8;i++) p[i] = fast_exp2_(fmaf(r[i], LOG2E_, -mxL));
    float s01 = p[0]+p[1], s23 = p[2]+p[3], s45 = p[4]+p[5], s67 = p[6]+p[7];
    float ts = (s01+s23) + (s45+s67);
    ts += __shfl_xor(ts, 16, 32);
    sm += ts;

    // ---- att^T += V^T_tile @ P^T (A pairs are adjacent slots of the loaded b128s) ----
    #pragma unroll
    for (int s=0;s<4;s++){
        v2f Pb = (v2f){p[2*s], p[2*s+1]};
        v2f A0 = (v2f){va[(s*3+0)*2], va[(s*3+0)*2+1]};
        v2f A1 = (v2f){va[(s*3+1)*2], va[(s*3+1)*2+1]};
        v2f A2 = (v2f){va[(s*3+2)*2], va[(s*3+2)*2+1]};
        acc0 = __builtin_amdgcn_wmma_f32_16x16x4_f32(false, A0, false, Pb, (short)0, acc0, false, false);
        acc1 = __builtin_amdgcn_wmma_f32_16x16x4_f32(false, A1, false, Pb, (short)0, acc1, false, false);
        acc2 = __builtin_amdgcn_wmma_f32_16x16x4_f32(false, A2, false, Pb, (short)0, acc2, false, false);
    }
}

// ------------- flash attention partial: one wave = 16 batch rows x 512 memory rows -----------
__global__ __launch_bounds__(32)
void attn_partial(const float* __restrict__ q, const float* __restrict__ Kswz,
                  const float* __restrict__ Vswz,
                  float* __restrict__ att_part, float* __restrict__ mx_part,
                  float* __restrict__ sm_part)
{
    const int chunk = blockIdx.x;
    const int b0 = blockIdx.y * 16;
    const int lane = threadIdx.x;
    const int col  = lane & 15;
    const int half = lane >> 4;

    // persistent B operand of the scores GEMM: q^T, B[k=d][n=b]
    v2f qb[12];
    {
        const float* qrow = q + (b0 + col)*D_ + 2*half;
        #pragma unroll
        for (int s=0;s<12;s++){
            float2 t = *(const float2*)(qrow + 4*s);
            qb[s] = (v2f){t.x, t.y};
        }
    }

    v8f acc0 = {0,0,0,0,0,0,0,0}, acc1 = acc0, acc2 = acc0;  // att^T tiles d0=0/16/32
    float mx = -1.0e30f, mxL = -1.0e30f * LOG2E_, sm = 0.0f;

    const size_t base0 = (size_t)(chunk*TILES)*768 + lane*24;
    const float* kb = Kswz + base0;
    const float* vb = Vswz + base0;

    float ka0[24], va0[24], ka1[24], va1[24];
    attn_tile_load(kb, vb, ka0, va0);
    for (int tile=0; tile<TILES; tile+=2){
        // prefetch tile+1 (always valid: TILES is even)
        attn_tile_load(kb + 768, vb + 768, ka1, va1);
        attn_tile_compute(ka0, va0, qb, acc0, acc1, acc2, mx, mxL, sm, half);
        // prefetch tile+2 (uniform clamp on the tail -> harmless redundant load)
        size_t step2 = (tile+2 < TILES) ? 1536 : 768;
        attn_tile_load(kb + step2, vb + step2, ka0, va0);
        attn_tile_compute(ka1, va1, qb, acc0, acc1, acc2, mx, mxL, sm, half);
        kb += 1536; vb += 1536;
    }

    // ---- write partials: lane owns batch row b0+col, 8 consecutive d per tile -> b128 stores ----
    {
        float* bp = att_part + ((size_t)chunk*B_ + b0 + col)*D_ + 8*half;
        float4 f;
        f.x=acc0[0]; f.y=acc0[1]; f.z=acc0[2]; f.w=acc0[3]; *(float4*)(bp +  0) = f;
        f.x=acc0[4]; f.y=acc0[5]; f.z=acc0[6]; f.w=acc0[7]; *(float4*)(bp +  4) = f;
        f.x=acc1[0]; f.y=acc1[1]; f.z=acc1[2]; f.w=acc1[3]; *(float4*)(bp + 16) = f;
        f.x=acc1[4]; f.y=acc1[5]; f.z=acc1[6]; f.w=acc1[7]; *(float4*)(bp + 20) = f;
        f.x=acc2[0]; f.y=acc2[1]; f.z=acc2[2]; f.w=acc2[3]; *(float4*)(bp + 32) = f;
        f.x=acc2[4]; f.y=acc2[5]; f.z=acc2[6]; f.w=acc2[7]; *(float4*)(bp + 36) = f;
    }
    if (half == 0){
        mx_part[chunk*B_ + b0 + col] = mx;
        sm_part[chunk*B_ + b0 + col] = sm;
    }
}

// ---------------- merge 64 chunk partials (log-sum-exp), c = q + att, store pred --------------
__global__ __launch_bounds__(32)
void combine_kernel(const float* __restrict__ att_part, const float* __restrict__ mx_part,
                    const float* __restrict__ sm_part, const float* __restrict__ q,
                    float* __restrict__ cstate, float* __restrict__ pred, int it)
{
    int b = blockIdx.x, lane = threadIdx.x;
    __shared__ float wf[NCHUNK];
    __shared__ float SMs;
    float mx0 = mx_part[lane*B_ + b],      sm0 = sm_part[lane*B_ + b];
    float mx1 = mx_part[(lane+32)*B_ + b], sm1 = sm_part[(lane+32)*B_ + b];
    float MX = fmaxf(mx0, mx1);
    #pragma unroll
    for (int o=16;o>0;o>>=1) MX = fmaxf(MX, __shfl_xor(MX, o, 32));
    float w0 = __expf(mx0 - MX), w1 = __expf(mx1 - MX);
    float sw = sm0*w0 + sm1*w1;
    #pragma unroll
    for (int o=16;o>0;o>>=1) sw += __shfl_xor(sw, o, 32);
    wf[lane] = w0; wf[lane+32] = w1;
    if (lane==0) SMs = sw;
    __syncthreads();
    float inv = 1.0f / SMs;
    for (int d=lane; d<D_; d+=32){
        float a = 0.0f;
        for (int c=0;c<NCHUNK;c++)
            a += att_part[((size_t)c*B_ + b)*D_ + d] * wf[c];
        a *= inv;
        cstate[b*D_+d] = q[b*D_+d] + a;
        pred[(size_t)(b*NPRED_ + it)*D_ + d] = a;
    }
}

// ---------------- decoder GRU(96) step; input only at t==0 ----------------
__global__ void dec_step(const float* __restrict__ h_in, float* __restrict__ h_out,
                         const float* __restrict__ sp, const float* __restrict__ pred,
                         const float* __restrict__ w_ih, const float* __restrict__ w_hh,
                         const float* __restrict__ b_ih, const float* __restrict__ b_hh,
                         int t)
{
    int idx = blockIdx.x*blockDim.x + threadIdx.x;
    if (idx >= NROW*96) return;
    int r = idx / 96, j = idx % 96;
    int b = r / NPRED_;
    float gi[3], gh[3];
    #pragma unroll
    for (int g=0; g<3; ++g){ gi[g] = b_ih[g*96+j]; gh[g] = b_hh[g*96+j]; }
    if (t == 0){
        const float* spr = sp + b*D_;
        const float* pr  = pred + (size_t)r*D_;
        #pragma unroll
        for (int g=0; g<3; ++g){
            const float* wi = w_ih + (size_t)(g*96+j)*96;
            float s = 0.0f;
            for (int k=0;k<D_;k++) s += spr[k]*wi[k];
            for (int k=0;k<D_;k++) s += pr[k]*wi[D_+k];
            gi[g] += s;
        }
    } else {
        const float* hr = h_in + (size_t)r*96;
        #pragma unroll
        for (int g=0; g<3; ++g){
            const float* wh = w_hh + (size_t)(g*96+j)*96;
            float s = 0.0f;
            for (int k=0;k<96;k++) s += hr[k]*wh[k];
            gh[g] += s;
        }
    }
    float rg = sigmoidf_(gi[0]+gh[0]);
    float z  = sigmoidf_(gi[1]+gh[1]);
    float nn = tanhf(gi[2] + rg*gh[2]);
    float hp = (t==0) ? 0.0f : h_in[(size_t)r*96 + j];
    h_out[(size_t)r*96 + j] = (1.0f - z)*nn + z*hp;
}

// ---------------- pres += fc(h); write coords ----------------
__global__ void dec_out(const float* __restrict__ h, const float* __restrict__ fc_w,
                        const float* __restrict__ fc_b, const float* __restrict__ obs,
                        float* __restrict__ pres, float* __restrict__ out, int t)
{
    int idx = blockIdx.x*blockDim.x + threadIdx.x;
    if (idx >= NROW*2) return;
    int r = idx >> 1, c = idx & 1;
    const float* hr = h + (size_t)r*96;
    const float* w = fc_w + c*96;
    float s = fc_b[c];
    for (int k=0;k<96;k++) s += hr[k]*w[k];
    float base = (t==0) ? obs[(r/NPRED_)*PAST_*2 + (PAST_-1)*2 + c] : pres[idx];
    float pv = base + s;
    pres[idx] = pv;
    out[(size_t)(r*FUT_ + t)*2 + c] = pv;
}

extern "C" void kernel_launch(void* const* d_in, const int* in_sizes, int n_in,
                              void* d_out, int out_size, void* d_ws, size_t ws_size,
                              hipStream_t stream)
{
    const float* past     = (const float*)d_in[0];
    const float* obs      = (const float*)d_in[1];
    const float* conv_w   = (const float*)d_in[2];
    const float* conv_b   = (const float*)d_in[3];
    const float* enc_w_ih = (const float*)d_in[4];
    const float* enc_w_hh = (const float*)d_in[5];
    const float* enc_b_ih = (const float*)d_in[6];
    const float* enc_b_hh = (const float*)d_in[7];
    const float* mem_p    = (const float*)d_in[8];
    const float* mem_f    = (const float*)d_in[9];
    const float* Wq = (const float*)d_in[10]; const float* bq = (const float*)d_in[11];
    const float* Wk = (const float*)d_in[12]; const float* bk = (const float*)d_in[13];
    const float* Wv = (const float*)d_in[14]; const float* bv = (const float*)d_in[15];
    const float* dw_ih = (const float*)d_in[16]; const float* dw_hh = (const float*)d_in[17];
    const float* db_ih = (const float*)d_in[18]; const float* db_hh = (const float*)d_in[19];
    const float* fc_w = (const float*)d_in[20]; const float* fc_b = (const float*)d_in[21];
    float* out = (float*)d_out;

    float* ws = (float*)d_ws;
    size_t off = 0;
    float* Kswz     = ws + off; off += (size_t)M_*D_;
    float* Vswz     = ws + off; off += (size_t)M_*D_;
    float* sp       = ws + off; off += (size_t)B_*D_;
    float* cst      = ws + off; off += (size_t)B_*D_;
    float* qb       = ws + off; off += (size_t)B_*D_;
    float* pred     = ws + off; off += (size_t)B_*NPRED_*D_;
    float* att_part = ws + off; off += (size_t)NCHUNK*B_*D_;
    float* mx_part  = ws + off; off += (size_t)NCHUNK*B_;
    float* sm_part  = ws + off; off += (size_t)NCHUNK*B_;
    float* h0       = ws + off; off += (size_t)NROW*96;
    float* h1       = ws + off; off += (size_t)NROW*96;
    float* pres     = ws + off; off += (size_t)NROW*2;

    enc_kernel<<<B_, 64, 0, stream>>>(past, conv_w, conv_b, enc_w_ih, enc_w_hh,
                                      enc_b_ih, enc_b_hh, sp);
    kv_kernel<<<(M_*D_ + 255)/256, 256, 0, stream>>>(mem_p, mem_f, Wk, bk, Wv, bv, Kswz, Vswz);

    for (int it=0; it<NPRED_; ++it){
        const float* cin = (it==0) ? sp : cst;
        q_kernel<<<(B_*D_ + 255)/256, 256, 0, stream>>>(cin, Wq, bq, qb);
        attn_partial<<<dim3(NCHUNK, B_/16), 32, 0, stream>>>(qb, Kswz, Vswz, att_part, mx_part, sm_part);
        combine_kernel<<<B_, 32, 0, stream>>>(att_part, mx_part, sm_part, qb, cst, pred, it);
    }

    float* hb[2] = {h0, h1};
    for (int t=0; t<FUT_; ++t){
        dec_step<<<(NROW*96 + 255)/256, 256, 0, stream>>>(hb[t&1], hb[(t+1)&1], sp, pred,
                                                          dw_ih, dw_hh, db_ih, db_hh, t);
        dec_out<<<(NROW*2 + 255)/256, 256, 0, stream>>>(hb[(t+1)&1], fc_w, fc_b, obs,
                                                        pres, out, t);
    }
}